// Attn_55448027792086
// MI455X (gfx1250) — compile-verified
//
#include <hip/hip_runtime.h>
#include <hip/hip_bf16.h>
#include <math.h>

typedef __attribute__((ext_vector_type(2))) float v2f;
typedef __attribute__((ext_vector_type(8))) float v8f;

#define S_LEN 2048
#define B_SZ  16
#define H_SZ  1024

// ---------------------------------------------------------------------------
// Phase 1: v[b,h] = sum_o hidden[b,o] * W[o,h]
// GEMM M=16 (b), N=1024 (h), K=1024 (o) using V_WMMA_F32_16X16X4_F32.
// One wave per 16-wide N tile -> 64 waves total.
// A(16x4) lane layout: a[j] = A[lane%16][k0 + 2*(lane/16) + j]
// B(4x16) lane layout: b[j] = B[k0 + 2*(lane/16) + j][n0 + lane%16]
// D(16x16) lane layout: d[r] = D[(lane/16)*8 + r][n0 + lane%16]
// ---------------------------------------------------------------------------
__global__ void hiddenW_wmma_kernel(const float* __restrict__ hidden,
                                    const float* __restrict__ W,
                                    float* __restrict__ v)
{
    const int wave = (blockIdx.x * blockDim.x + threadIdx.x) >> 5;  // 0..63
    const int lane = threadIdx.x & 31;
    const int n0   = wave * 16;
    const int m    = lane & 15;          // A row (= b index)
    const int kh   = (lane >> 4) * 2;    // k sub-offset: 0 or 2
    const int col  = n0 + (lane & 15);   // B/D column (= h index)

    v8f acc = {};
    #pragma unroll 4
    for (int k0 = 0; k0 < H_SZ; k0 += 4) {
        v2f a, b;
        // hidden is [1,B,H] -> hidden[m*H + o]; two consecutive K values
        const float2 av = *(const float2*)(hidden + m * H_SZ + k0 + kh);
        a.x = av.x; a.y = av.y;
        // W is [H_out=o, H_in=h] row-major; B[k][n] = W[k*H + n]
        b.x = W[(size_t)(k0 + kh + 0) * H_SZ + col];
        b.y = W[(size_t)(k0 + kh + 1) * H_SZ + col];
        acc = __builtin_amdgcn_wmma_f32_16x16x4_f32(
            /*neg_a=*/false, a, /*neg_b=*/false, b,
            /*c_mod=*/(short)0, acc, /*reuse_a=*/false, /*reuse_b=*/false);
    }

    const int mrow = (lane >> 4) * 8;
    #pragma unroll
    for (int r = 0; r < 8; ++r) {
        v[(mrow + r) * H_SZ + col] = acc[r];
    }
}

// ---------------------------------------------------------------------------
// Phase 2: energies[b,s] = v[b,:] . enc[s,b,:]
// One wave per (s,b) pair; each wave streams one contiguous 4KB row of enc
// with float4 loads (coalesced 512B per wave-load). v rows are L2-resident.
// Memory-bound: streams the 134MB encoder tensor exactly once.
// ---------------------------------------------------------------------------
__global__ void energies_kernel(const float* __restrict__ enc,
                                const float* __restrict__ v,
                                float* __restrict__ E)
{
    const int wavesPerBlock = blockDim.x >> 5;
    const int wave = blockIdx.x * wavesPerBlock + (threadIdx.x >> 5); // 0..32767
    const int lane = threadIdx.x & 31;
    const int s = wave >> 4;
    const int b = wave & 15;

    const float4* __restrict__ ep =
        (const float4*)(enc + ((size_t)s * B_SZ + b) * H_SZ);
    const float4* __restrict__ vp =
        (const float4*)(v + (size_t)b * H_SZ);

    float acc = 0.f;
    #pragma unroll
    for (int i = 0; i < 8; ++i) {               // 8 * 32 lanes * 4 = 1024
        const float4 e = ep[i * 32 + lane];
        const float4 w = vp[i * 32 + lane];
        acc += e.x * w.x + e.y * w.y + e.z * w.z + e.w * w.w;
    }
    // wave32 cross-lane reduction
    #pragma unroll
    for (int off = 16; off > 0; off >>= 1)
        acc += __shfl_xor(acc, off, 32);

    if (lane == 0)
        E[b * S_LEN + s] = acc;
}

// ---------------------------------------------------------------------------
// Phase 3: row softmax over S=2048. One block (256 thr) per batch row.
// Note: the hidden.bias term is constant per row and cancels in softmax.
// ---------------------------------------------------------------------------
__global__ void softmax_kernel(const float* __restrict__ E,
                               float* __restrict__ out)
{
    const int b   = blockIdx.x;
    const int tid = threadIdx.x;     // 0..255
    __shared__ float red[256];

    float vals[8];
    float mx = -INFINITY;
    #pragma unroll
    for (int i = 0; i < 8; ++i) {
        vals[i] = E[b * S_LEN + i * 256 + tid];
        mx = fmaxf(mx, vals[i]);
    }
    red[tid] = mx;
    __syncthreads();
    for (int st = 128; st > 0; st >>= 1) {
        if (tid < st) red[tid] = fmaxf(red[tid], red[tid + st]);
        __syncthreads();
    }
    mx = red[0];
    __syncthreads();

    float sum = 0.f;
    #pragma unroll
    for (int i = 0; i < 8; ++i) {
        vals[i] = __expf(vals[i] - mx);
        sum += vals[i];
    }
    red[tid] = sum;
    __syncthreads();
    for (int st = 128; st > 0; st >>= 1) {
        if (tid < st) red[tid] += red[tid + st];
        __syncthreads();
    }
    const float inv = 1.0f / red[0];

    #pragma unroll
    for (int i = 0; i < 8; ++i)
        out[b * S_LEN + i * 256 + tid] = vals[i] * inv;
}

extern "C" void kernel_launch(void* const* d_in, const int* in_sizes, int n_in,
                              void* d_out, int out_size, void* d_ws, size_t ws_size,
                              hipStream_t stream)
{
    const float* hidden = (const float*)d_in[0];  // [1, 16, 1024]
    const float* enc    = (const float*)d_in[1];  // [2048, 16, 1024]
    const float* W      = (const float*)d_in[2];  // [1024, 1024]
    // d_in[3] = bias: cancels under softmax, unused.
    float* out = (float*)d_out;                   // [16, 1, 2048]

    float* v = (float*)d_ws;                      // 16*1024 floats = 64 KB
    float* E = v + B_SZ * H_SZ;                   // 16*2048 floats = 128 KB

    // Phase 1: 64 waves = 8 blocks x 256 threads (8 waves each)
    hiddenW_wmma_kernel<<<8, 256, 0, stream>>>(hidden, W, v);

    // Phase 2: 32768 waves = 4096 blocks x 256 threads
    energies_kernel<<<4096, 256, 0, stream>>>(enc, v, E);

    // Phase 3: one block per batch row
    softmax_kernel<<<B_SZ, 256, 0, stream>>>(E, out);
}